// mfaLayer_53008486367980
// MI455X (gfx1250) — compile-verified
//
#include <hip/hip_runtime.h>

#define NN 1024
#define EPSF 1e-16f

typedef __attribute__((ext_vector_type(2))) float v2f;
typedef __attribute__((ext_vector_type(8))) float v8f;
typedef __attribute__((ext_vector_type(4))) unsigned int v4u;
typedef __attribute__((ext_vector_type(8))) int v8i;
typedef __attribute__((ext_vector_type(4))) int v4i;

// Fast hardware transcendentals (no IEEE fixup chains).
__device__ __forceinline__ float fast_ln(float x)  { return 0.69314718056f * __builtin_amdgcn_logf(x); }
__device__ __forceinline__ float fast_exp(float x) { return __builtin_amdgcn_exp2f(1.44269504089f * x); }
__device__ __forceinline__ float fast_rcp(float x) { return __builtin_amdgcn_rcpf(x); }

// ------------------------------------------------------------------
// Tensor Data Mover: DMA a 2D tile of f32 from X (1024x1024, row stride
// 1024 elements) into LDS.  D# built per CDNA5 ISA.
// Issue from ONE wave only; pair with s_wait_tensorcnt(0) + barrier.
// This toolchain exposes the 6-arg builtin:
//   (uint32x4 g0, int32x8 g1, int32x4 g2, int32x4 g3, int32x8 pad, i32 cpol)
// ------------------------------------------------------------------
__device__ __forceinline__ void tdm_load_tile_f32(const float* gsrc, void* lds_dst,
                                                  unsigned tile_w, unsigned tile_h)
{
    const unsigned lds_off = (unsigned)(unsigned long long)(uintptr_t)lds_dst; // LDS byte offset
    const unsigned long long ga = (unsigned long long)(uintptr_t)gsrc;

    v4u g0;
    g0[0] = 1u;                                            // count=1, user D#, no gather
    g0[1] = lds_off;                                       // lds_addr
    g0[2] = (unsigned)(ga & 0xFFFFFFFFu);                  // global_addr[31:0]
    g0[3] = (unsigned)((ga >> 32) & 0x01FFFFFFu)           // global_addr[56:32]
          | (2u << 30);                                    // type = 2 ("image")

    v8i g1;
    g1[0] = 0x00020000;                                    // data_size=2 (4B); wg_mask=0
    g1[1] = (int)((NN & 0xFFFFu) << 16);                   // tensor_dim0[15:0] @ bits 63:48
    g1[2] = (int)((NN & 0xFFFFu) << 16);                   // dim0[31:16]=0 | tensor_dim1[15:0] @ 95:80
    g1[3] = (int)((tile_w & 0xFFFFu) << 16);               // dim1[31:16]=0 | tile_dim0 @ 127:112
    g1[4] = (int)(tile_h & 0xFFFFu);                       // tile_dim1 @ 143:128 ; tile_dim2=0
    g1[5] = (int)NN;                                       // tensor_dim0_stride[31:0]
    g1[6] = 0;                                             // stride hi / dim1_stride (unused, 2D)
    g1[7] = 0;

    v4i gz4 = {0, 0, 0, 0};                                // groups 2/3 unused (<=2D tensor)
    v8i gz8 = {0, 0, 0, 0, 0, 0, 0, 0};
    __builtin_amdgcn_tensor_load_to_lds(g0, g1, gz4, gz4, gz8, 0);
}

// ------------------------------------------------------------------
// Kernel 1: per-row stats. One block = 16 rows, 256 threads = 8 waves.
// Row sums of log(1-x+eps) and x/(1-x+eps) on the matrix pipe:
//   D(16x16) = A(16x4) x ones(4x16) + C  -> row sums replicated per column.
// ------------------------------------------------------------------
__global__ __launch_bounds__(256) void row_stats_kernel(
    const float* __restrict__ X,
    const int*   __restrict__ iT,          // (N,21,2)
    const float* __restrict__ b0p, const float* __restrict__ b1p,
    const float* __restrict__ b2p,
    float* __restrict__ row_lsum,          // ws[0..1023]
    float* __restrict__ part)              // ws[1024 + blockIdx]
{
    __shared__ float ts[16 * 512];         // 32 KB half-tile (16 rows x 512 cols)
    __shared__ float partL[8 * 16];
    __shared__ float partX[8 * 16];
    __shared__ float rowred[16];

    const int tid  = threadIdx.x;
    const int lane = tid & 31;             // wave32
    const int wave = tid >> 5;
    const int i0   = blockIdx.x * 16;

    const int m  = lane & 15;              // A-fragment row (M)
    const int kb = (lane >> 4) << 1;       // K sub-offset: lanes 0-15 -> 0, lanes 16-31 -> 2

    v8f accL = {};                         // row sums of log(1-x+eps)
    v8f accX = {};                         // row sums of x/(1-x+eps)
    v2f bones; bones.x = 1.0f; bones.y = 1.0f;   // B = all-ones

    for (int h = 0; h < 2; ++h) {
        // async TDM copy of the 16x512 tile into LDS (wave 0 issues + waits)
        if (wave == 0) {
            tdm_load_tile_f32(X + (size_t)i0 * NN + h * 512, ts, 512, 16);
            __builtin_amdgcn_s_wait_tensorcnt(0);
        }
        __syncthreads();

        const int cbase = wave * 64;       // each wave owns 64 columns of this half
        for (int s = 0; s < 64; s += 4) {
            int k0 = cbase + s + kb;
            float x0 = ts[m * 512 + k0];
            float x1 = ts[m * 512 + k0 + 1];
            float d0 = 1.0f - x0 + EPSF;
            float d1 = 1.0f - x1 + EPSF;
            v2f aL, aX;
            aL.x = fast_ln(d0);       aL.y = fast_ln(d1);
            aX.x = x0 * fast_rcp(d0); aX.y = x1 * fast_rcp(d1);
            accL = __builtin_amdgcn_wmma_f32_16x16x4_f32(false, aL, false, bones,
                                                         (short)0, accL, false, false);
            accX = __builtin_amdgcn_wmma_f32_16x16x4_f32(false, aX, false, bones,
                                                         (short)0, accX, false, false);
        }
        __syncthreads();
    }

    // D layout: VGPR r holds M=r (lanes 0-15) and M=8+r (lanes 16-31)
    if (lane == 0) {
#pragma unroll
        for (int r = 0; r < 8; ++r) { partL[wave * 16 + r] = accL[r]; partX[wave * 16 + r] = accX[r]; }
    } else if (lane == 16) {
#pragma unroll
        for (int r = 0; r < 8; ++r) { partL[wave * 16 + 8 + r] = accL[r]; partX[wave * 16 + 8 + r] = accX[r]; }
    }
    __syncthreads();

    if (tid < 16) {
        float rl = 0.f, sx = 0.f;
#pragma unroll
        for (int w = 0; w < 8; ++w) { rl += partL[w * 16 + tid]; sx += partX[w * 16 + tid]; }
        const int i = i0 + tid;
        row_lsum[i] = rl;

        // sum_c prod_e XX_[i, iT[i,c,e]]  (21 pairs, L2-resident X)
        float prod = 0.f;
        for (int c = 0; c < 21; ++c) {
            int j0 = iT[((size_t)i * 21 + c) * 2 + 0];
            int j1 = iT[((size_t)i * 21 + c) * 2 + 1];
            float xa = X[(size_t)i * NN + j0];
            float xb = X[(size_t)i * NN + j1];
            float xxa = xa * fast_rcp(1.0f - xa + EPSF);
            float xxb = xb * fast_rcp(1.0f - xb + EPSF);
            prod += xxa * xxb;
        }
        rowred[tid] = (b0p[0] + sx * b1p[0] + prod * b2p[0]) * fast_exp(rl);
    }
    __syncthreads();
    if (tid == 0) {
        float s = 0.f;
#pragma unroll
        for (int r = 0; r < 16; ++r) s += rowred[r];
        part[blockIdx.x] = s;
    }
}

// ------------------------------------------------------------------
// Kernel 2: main gamma + elb terms. One block per row i.
// ------------------------------------------------------------------
__global__ __launch_bounds__(256) void mfa_main_kernel(
    const float* __restrict__ X,
    const float* __restrict__ C, const float* __restrict__ D,
    const float* __restrict__ S, const float* __restrict__ Mb,
    const float* __restrict__ Tp,
    const float* __restrict__ b0p, const float* __restrict__ b1p, const float* __restrict__ b2p,
    const float* __restrict__ lamp, const float* __restrict__ etap,
    const float* __restrict__ nup,  const float* __restrict__ mup,
    const float* __restrict__ kapp,
    const int* __restrict__ ij, const int* __restrict__ ijF, const int* __restrict__ ijT,
    const float* __restrict__ row_lsum,
    float* __restrict__ outg,              // d_out + 1
    float* __restrict__ part)              // ws + 1024 + 64
{
    __shared__ float xr[NN], lxr[NN], xxr[NN];
    __shared__ float red[256];

    const int tid = threadIdx.x;
    const int i   = blockIdx.x;

    // async TDM copy of row i (4 KB, 1024x1 tile) into LDS
    if (tid < 32) {
        tdm_load_tile_f32(X + (size_t)i * NN, xr, 1024, 1);
        __builtin_amdgcn_s_wait_tensorcnt(0);
    }
    __syncthreads();

    // derive log(1-x+eps) and x/(1-x+eps) rows from the staged x row
    {
        float4 xv = *(const float4*)(xr + tid * 4);
        float4 lv, qv;
        float* xs = (float*)&xv; float* ls = (float*)&lv; float* qs = (float*)&qv;
#pragma unroll
        for (int j = 0; j < 4; ++j) {
            float dd = 1.0f - xs[j] + EPSF;
            ls[j] = fast_ln(dd);
            qs[j] = xs[j] * fast_rcp(dd);
        }
        *(float4*)(lxr + tid * 4) = lv;
        *(float4*)(xxr + tid * 4) = qv;
    }
    __syncthreads();

    const float t   = Tp[0];
    const float b0  = b0p[0],  b1 = b1p[0], b2 = b2p[0];
    const float lam = lamp[0], eta = etap[0], nu = nup[0], mu = mup[0];
    float kapv[6];
#pragma unroll
    for (int d = 0; d < 6; ++d) kapv[d] = kapp[d];
    const float rl   = row_lsum[i];
    const float invt = fast_rcp(t);

    float acc = 0.0f;

    for (int j = 0; j < 4; ++j) {
        const int    k    = tid + j * 256;           // coalesced across the block
        const size_t base = (size_t)i * NN + k;
        const int* pij = ij  + base * 6;
        const int* pf  = ijF + base * 6;
        const int* pt  = ijT + base * 30;
        if (j < 3) __builtin_prefetch((const void*)(ijT + (base + 256) * 30), 0, 1);

        const float x  = xr[k];
        const float x_ = 1.0f - x + EPSF;
        const float lx = lxr[k];

        float locUp0 = 0.f, kapt = 0.f, t3 = 0.f;
#pragma unroll
        for (int d = 0; d < 6; ++d) {
            const int   g  = pij[d];
            const float xg = xr[g];
            locUp0 += xxr[g];
            const float kd = kapv[d];
            kapt += (4.0f * xg - 2.0f) * kd;
            t3   += (1.0f - 2.0f * (xg + x) + 4.0f * xg * x) * kd;
        }
        float locUp1 = 0.f;
#pragma unroll
        for (int c = 0; c < 6; ++c) {
            float s5 = 0.f;
#pragma unroll
            for (int m5 = 0; m5 < 5; ++m5) s5 += xxr[pt[c * 5 + m5]];
            locUp1 += xxr[pf[c]] * s5;
        }
        const float pr = lam + eta * D[base] + nu * C[base] + mu * S[base];
        const float mb = Mb[base];
        const float xt = X[(size_t)k * NN + i];      // transpose read, L2-resident
        const float tmp0 = fast_exp(rl - lx);

        float gamma = (b1 - b0 + (b2 - b1) * locUp0 - b2 * locUp1) * tmp0
                    + (4.0f * xt - 2.0f) * pr + 100.0f * (mb - 1.0f) + kapt;
        outg[base] = fast_rcp(1.0f + fast_exp(-gamma * invt));

        acc += pr * (1.0f + 4.0f * x * xt - 2.0f * (x + xt)) * mb
             + t3 * mb
             - t * (x * fast_ln(x + EPSF) + x_ * lx) * mb;
    }

    red[tid] = acc;
    __syncthreads();
    for (int off = 128; off > 0; off >>= 1) {
        if (tid < off) red[tid] += red[tid + off];
        __syncthreads();
    }
    if (tid == 0) part[i] = red[0];
}

// ------------------------------------------------------------------
// Kernel 3: deterministic final reduction of 64 + 1024 partials.
// ------------------------------------------------------------------
__global__ __launch_bounds__(256) void reduce_kernel(
    const float* __restrict__ part, int n, float* __restrict__ out)
{
    __shared__ float red[256];
    float s = 0.f;
    for (int idx = threadIdx.x; idx < n; idx += 256) s += part[idx];
    red[threadIdx.x] = s;
    __syncthreads();
    for (int off = 128; off > 0; off >>= 1) {
        if (threadIdx.x < off) red[threadIdx.x] += red[threadIdx.x + off];
        __syncthreads();
    }
    if (threadIdx.x == 0) out[0] = red[0];
}

extern "C" void kernel_launch(void* const* d_in, const int* in_sizes, int n_in,
                              void* d_out, int out_size, void* d_ws, size_t ws_size,
                              hipStream_t stream) {
    const float* X   = (const float*)d_in[0];
    const float* T   = (const float*)d_in[1];
    const float* C   = (const float*)d_in[2];
    const float* D   = (const float*)d_in[3];
    const float* S   = (const float*)d_in[4];
    const float* Mk  = (const float*)d_in[5];
    const float* b0  = (const float*)d_in[6];
    const float* b1  = (const float*)d_in[7];
    const float* b2  = (const float*)d_in[8];
    const float* lam = (const float*)d_in[9];
    const float* eta = (const float*)d_in[10];
    const float* nu  = (const float*)d_in[11];
    const float* mu  = (const float*)d_in[12];
    const float* kap = (const float*)d_in[13];
    const int* ij  = (const int*)d_in[14];
    const int* ijF = (const int*)d_in[15];
    const int* ijT = (const int*)d_in[16];
    const int* iT  = (const int*)d_in[17];
    // d_in[18] = b (always 0; trailing dims are size 1)

    float* ws       = (float*)d_ws;
    float* rowlsum  = ws;              // [0..1023]
    float* parts    = ws + NN;         // [0..63] row_stats, [64..1087] main

    float* out = (float*)d_out;        // out[0] = elb ; out[1..] = sigmoid(gamma/t)

    row_stats_kernel<<<NN / 16, 256, 0, stream>>>(X, iT, b0, b1, b2, rowlsum, parts);
    mfa_main_kernel<<<NN, 256, 0, stream>>>(X, C, D, S, Mk, T, b0, b1, b2,
                                            lam, eta, nu, mu, kap,
                                            ij, ijF, ijT, rowlsum, out + 1, parts + 64);
    reduce_kernel<<<1, 256, 0, stream>>>(parts, 64 + NN, out);
}